// RNNCellCmplx_6047313953020
// MI455X (gfx1250) — compile-verified
//
#include <hip/hip_runtime.h>
#include <hip/hip_bf16.h>

typedef __attribute__((ext_vector_type(2))) float v2f;
typedef __attribute__((ext_vector_type(8))) float v8f;

#define BATCH   128
#define INDIM   256
#define HDIM    1024
#define NLAYER  256   // 128 A-layers interleaved with 128 B-layers
#define NA      512   // pairs per A layer
#define NB      511   // pairs per B layer

// ---------------------------------------------------------------------------
// Kernel 1: precompute cos/sin of every mesh angle once.
// Table layout: layer j in [0,256), slot p in [0,512): (cos th0, sin th0,
// cos th1, sin th1). Even j = A-layer l=j/2, odd j = B-layer l=j/2.
// 2 MB total -> stays L2-resident for all 128 mesh workgroups.
// ---------------------------------------------------------------------------
__global__ __launch_bounds__(512) void trig_kernel(
    const float* __restrict__ A0, const float* __restrict__ A1,
    const float* __restrict__ B0, const float* __restrict__ B1,
    float* __restrict__ table) {
  int j = blockIdx.x;   // layer 0..255
  int p = threadIdx.x;  // pair  0..511
  int l = j >> 1;
  float th0 = 0.f, th1 = 0.f;
  if ((j & 1) == 0) {
    th0 = A0[l * NA + p];
    th1 = A1[l * NA + p];
  } else if (p < NB) {
    th0 = B0[l * NB + p];
    th1 = B1[l * NB + p];
  }
  float s0, c0, s1, c1;
  sincosf(th0, &s0, &c0);
  sincosf(th1, &s1, &c1);
  float4 v = make_float4(c0, s0, c1, s1);
  *(float4*)(table + ((size_t)j * NA + p) * 4) = v;
}

// ---------------------------------------------------------------------------
// Kernel 2: complex CLinear via V_WMMA_F32_16X16X4_F32.
//   ihR = inR @ wR^T - inI @ wI^T + bR
//   ihI = inR @ wI^T + inI @ wR^T + bI
// One 16x16 output tile per wave; 4 f32 accumulators (RR, II, RI, IR) since
// f32 WMMA has no A/B negate modifier. A frag: lane(l16,half) holds
// A[tm+l16][k+2*half .. +1] (contiguous b64). B = weight^T so B[k][n] =
// weight[n][k], also a contiguous b64 per lane.
// ---------------------------------------------------------------------------
__global__ __launch_bounds__(256) void gemm_kernel(
    const float* __restrict__ inR, const float* __restrict__ inI,
    const float* __restrict__ wR,  const float* __restrict__ wI,
    const float* __restrict__ bR,  const float* __restrict__ bI,
    float* __restrict__ ihR, float* __restrict__ ihI) {
  const int lane = threadIdx.x & 31;
  const int wid  = blockIdx.x * (blockDim.x >> 5) + (threadIdx.x >> 5);
  const int tm   = (wid & 7) * 16;   // 8 M-tiles  (BATCH/16)
  const int tn   = (wid >> 3) * 16;  // 64 N-tiles (HDIM/16)
  const int half = lane >> 4;
  const int l16  = lane & 15;

  const float* ar_p = inR + (size_t)(tm + l16) * INDIM + 2 * half;
  const float* ai_p = inI + (size_t)(tm + l16) * INDIM + 2 * half;
  const float* br_p = wR  + (size_t)(tn + l16) * INDIM + 2 * half;
  const float* bi_p = wI  + (size_t)(tn + l16) * INDIM + 2 * half;

  v8f prr = {}, pii = {}, pri = {}, pir = {};
#pragma unroll 4
  for (int k = 0; k < INDIM; k += 4) {
    v2f ar = *(const v2f*)(ar_p + k);
    v2f ai = *(const v2f*)(ai_p + k);
    v2f br = *(const v2f*)(br_p + k);
    v2f bi = *(const v2f*)(bi_p + k);
    prr = __builtin_amdgcn_wmma_f32_16x16x4_f32(false, ar, false, br, (short)0, prr, false, false);
    pii = __builtin_amdgcn_wmma_f32_16x16x4_f32(false, ai, false, bi, (short)0, pii, false, false);
    pri = __builtin_amdgcn_wmma_f32_16x16x4_f32(false, ar, false, bi, (short)0, pri, false, false);
    pir = __builtin_amdgcn_wmma_f32_16x16x4_f32(false, ai, false, br, (short)0, pir, false, false);
  }

  const int n = tn + l16;
  const float biasr = bR[n];
  const float biasi = bI[n];
#pragma unroll
  for (int r = 0; r < 8; ++r) {
    // C/D layout: VGPR r, lanes 0-15 -> M=r, lanes 16-31 -> M=8+r, N=lane%16
    const int m = tm + r + 8 * half;
    ihR[(size_t)m * HDIM + n] = prr[r] - pii[r] + biasr;
    ihI[(size_t)m * HDIM + n] = pri[r] + pir[r] + biasi;
  }
}

// ---------------------------------------------------------------------------
// Kernel 3: 256-layer Clements mesh with LDS-resident state, fused epilogue.
// One workgroup (512 threads) per batch row; thread t executes one MZI per
// layer (A layer: channels (2t,2t+1); B layer: (2t+1,2t+2), t<511).
// ---------------------------------------------------------------------------
__global__ __launch_bounds__(512) void mesh_kernel(
    const float* __restrict__ stateR, const float* __restrict__ stateI,
    const float* __restrict__ table,
    const float* __restrict__ ihR, const float* __restrict__ ihI,
    const float* __restrict__ omega, const float* __restrict__ mod_bias,
    float* __restrict__ out) {
  __shared__ float xr[HDIM];
  __shared__ float xi[HDIM];

  const int b = blockIdx.x;
  const int t = threadIdx.x;

  // Load state (two channels per thread, coalesced b64)
  {
    const float2 sr = *(const float2*)(stateR + (size_t)b * HDIM + 2 * t);
    const float2 si = *(const float2*)(stateI + (size_t)b * HDIM + 2 * t);
    xr[2 * t] = sr.x; xr[2 * t + 1] = sr.y;
    xi[2 * t] = si.x; xi[2 * t + 1] = si.y;
  }
  __syncthreads();

  for (int j = 0; j < NLAYER; ++j) {
    const float4 tg = *(const float4*)(table + ((size_t)j * NA + t) * 4);
    const int  isB    = j & 1;
    const int  base   = 2 * t + isB;
    const bool active = !isB || (t < NB);
    if (active) {
      const float ar = xr[base],     ai = xi[base];
      const float br = xr[base + 1], bi = xi[base + 1];
      // phase shifter e^{i th0} on arm a
      const float pr = tg.x * ar - tg.y * ai;
      const float pq = tg.y * ar + tg.x * ai;
      // directional coupler, angle th1
      xr[base]     = tg.z * pr - tg.w * bi;
      xi[base]     = tg.z * pq + tg.w * br;
      xr[base + 1] = tg.z * br - tg.w * pq;
      xi[base + 1] = tg.z * bi + tg.w * pr;
    }
    __syncthreads();
  }

  // Epilogue: diagonal phase e^{i omega}, add ih, modReLU
#pragma unroll
  for (int q = 0; q < 2; ++q) {
    const int ch = 2 * t + q;
    float so, co;
    sincosf(omega[ch], &so, &co);
    const float hr = xr[ch], hi = xi[ch];
    const float hhR = co * hr - so * hi;
    const float hhI = so * hr + co * hi;
    const float zr = ihR[(size_t)b * HDIM + ch] + hhR;
    const float zi = ihI[(size_t)b * HDIM + ch] + hhI;
    const float mag = sqrtf(zr * zr + zi * zi);
    const float sc  = fmaxf(mag + mod_bias[ch], 0.f) / fmaxf(mag, 1e-8f);
    out[(size_t)b * HDIM + ch]                        = sc * zr;
    out[(size_t)BATCH * HDIM + (size_t)b * HDIM + ch] = sc * zi;
  }
}

// ---------------------------------------------------------------------------
extern "C" void kernel_launch(void* const* d_in, const int* in_sizes, int n_in,
                              void* d_out, int out_size, void* d_ws, size_t ws_size,
                              hipStream_t stream) {
  const float* inputsR = (const float*)d_in[0];
  const float* inputsI = (const float*)d_in[1];
  const float* stateR  = (const float*)d_in[2];
  const float* stateI  = (const float*)d_in[3];
  const float* weightR = (const float*)d_in[4];
  const float* weightI = (const float*)d_in[5];
  const float* biasR   = (const float*)d_in[6];
  const float* biasI   = (const float*)d_in[7];
  const float* angleA0 = (const float*)d_in[8];
  const float* angleA1 = (const float*)d_in[9];
  const float* angleB0 = (const float*)d_in[10];
  const float* angleB1 = (const float*)d_in[11];
  const float* omega   = (const float*)d_in[12];
  const float* modb    = (const float*)d_in[13];

  float* ws    = (float*)d_ws;
  float* ihR   = ws;                          // 128*1024 floats
  float* ihI   = ws + BATCH * HDIM;           // 128*1024 floats
  float* table = ws + 2 * BATCH * HDIM;       // 256*512*4 floats

  trig_kernel<<<NLAYER, 512, 0, stream>>>(angleA0, angleA1, angleB0, angleB1, table);
  gemm_kernel<<<64, 256, 0, stream>>>(inputsR, inputsI, weightR, weightI,
                                      biasR, biasI, ihR, ihI);
  mesh_kernel<<<BATCH, 512, 0, stream>>>(stateR, stateI, table, ihR, ihI,
                                         omega, modb, (float*)d_out);
}